// Graph_71854802862656
// MI455X (gfx1250) — compile-verified
//
#include <hip/hip_runtime.h>
#include <hip/hip_bf16.h>

typedef unsigned int u32;
typedef __attribute__((ext_vector_type(2))) float v2f;
typedef __attribute__((ext_vector_type(8))) float v8f;

#define EPS_ 1e-8f

__device__ __forceinline__ float sigmoidf_(float x) {
    return 1.0f / (1.0f + __expf(-x));
}

// ---------------- zero fill ----------------
__global__ void k_zero(u32* __restrict__ p, int n) {
    int i = blockIdx.x * blockDim.x + threadIdx.x;
    if (i < n) p[i] = 0u;
}

// ---------------- per-node norms (one wave32 per node) ----------------
__global__ void k_norms(const float* __restrict__ fa, const float* __restrict__ fb,
                        float* __restrict__ na, float* __restrict__ nb, int N) {
    int t = blockIdx.x * blockDim.x + threadIdx.x;
    int node = t >> 5;
    int lane = t & 31;
    if (node >= N) return;
    float x = fa[(size_t)node * 64 + lane];
    float y = fa[(size_t)node * 64 + 32 + lane];
    float sa = x * x + y * y;
    float z = fb[(size_t)node * 32 + lane];
    float sb = z * z;
    #pragma unroll
    for (int off = 16; off > 0; off >>= 1) {
        sa += __shfl_xor(sa, off, 32);
        sb += __shfl_xor(sb, off, 32);
    }
    if (lane == 0) {
        na[node] = sqrtf(sa);
        nb[node] = sqrtf(sb);
    }
}

// ---------------- WMMA edge scores: one wave handles 16 edges ----------------
// A = 16 gathered src rows, B = 16 gathered dst rows; diag(A x B^T) = per-edge dots.
// f32 16x16x4 WMMA, chained over K (16 steps for d=64, 8 steps for d=32).
__global__ void k_score(const float* __restrict__ fa, const float* __restrict__ fb,
                        const float* __restrict__ na, const float* __restrict__ nb,
                        const int* __restrict__ src, const int* __restrict__ dst,
                        float* __restrict__ outsa, float* __restrict__ outsb, int E) {
    int t = blockIdx.x * blockDim.x + threadIdx.x;
    int wave = t >> 5;
    int lane = t & 31;
    int ebase = wave * 16;
    if (ebase >= E) return;
    int half = lane >> 4;   // 0: K={0,1}+4k, 1: K={2,3}+4k  (A 16x4 f32 layout)
    int r = lane & 15;      // A row / B col -> edge-in-group index
    int e = ebase + r;
    if (e >= E) e = E - 1;
    int srow = src[e], drow = dst[e];
    // diag(i,i): i in 0..7 -> lane i, acc[i]; i in 8..15 -> lane 16+i? (N=lane-16=i -> lane=i+16), acc[i-8]
    bool have = ((half == 0) && (r < 8)) || ((half == 1) && (r >= 8));
    bool valid = have && ((ebase + r) < E);
    int ci = r & 7;

    // ---- modality A (d = 64) ----
    {
        const float* ap = fa + (size_t)srow * 64 + half * 2;
        const float* bp = fa + (size_t)drow * 64 + half * 2;
        v8f acc = {0.f, 0.f, 0.f, 0.f, 0.f, 0.f, 0.f, 0.f};
        #pragma unroll
        for (int kc = 0; kc < 16; ++kc) {
            v2f A = *(const v2f*)(ap + kc * 4);
            v2f B = *(const v2f*)(bp + kc * 4);
            acc = __builtin_amdgcn_wmma_f32_16x16x4_f32(
                false, A, false, B, (short)0, acc, false, false);
        }
        float dotv = 0.f;
        #pragma unroll
        for (int j = 0; j < 8; ++j)
            if (ci == j) dotv = acc[j];
        float nu = na[srow], nv = na[drow];
        float sc = sigmoidf_(sigmoidf_(dotv) / (nu * nv + EPS_));
        if (valid) outsa[e] = sc;
    }

    // ---- modality B (d = 32) ----
    {
        const float* ap = fb + (size_t)srow * 32 + half * 2;
        const float* bp = fb + (size_t)drow * 32 + half * 2;
        v8f acc = {0.f, 0.f, 0.f, 0.f, 0.f, 0.f, 0.f, 0.f};
        #pragma unroll
        for (int kc = 0; kc < 8; ++kc) {
            v2f A = *(const v2f*)(ap + kc * 4);
            v2f B = *(const v2f*)(bp + kc * 4);
            acc = __builtin_amdgcn_wmma_f32_16x16x4_f32(
                false, A, false, B, (short)0, acc, false, false);
        }
        float dotv = 0.f;
        #pragma unroll
        for (int j = 0; j < 8; ++j)
            if (ci == j) dotv = acc[j];
        float nu = nb[srow], nv = nb[drow];
        float sc = sigmoidf_(sigmoidf_(dotv) / (nu * nv + EPS_));
        if (valid) outsb[e] = sc;
    }
}

// ---------------- 3-level radix top-k selection ----------------
// Scores are positive floats => u32 bit order == float order.
__global__ void k_hist1(const float* __restrict__ s, u32* __restrict__ h, int E) {
    int e = blockIdx.x * blockDim.x + threadIdx.x;
    if (e >= E) return;
    u32 bits = __float_as_uint(s[e]);
    atomicAdd(&h[bits >> 20], 1u);
}

__global__ void k_hist2(const float* __restrict__ s, const u32* __restrict__ p,
                        u32* __restrict__ h, int E) {
    int e = blockIdx.x * blockDim.x + threadIdx.x;
    if (e >= E) return;
    u32 bits = __float_as_uint(s[e]);
    if ((bits >> 20) == p[0]) atomicAdd(&h[(bits >> 8) & 0xFFFu], 1u);
}

__global__ void k_hist3(const float* __restrict__ s, const u32* __restrict__ p,
                        u32* __restrict__ h, int E) {
    int e = blockIdx.x * blockDim.x + threadIdx.x;
    if (e >= E) return;
    u32 bits = __float_as_uint(s[e]);
    u32 pref = (p[0] << 12) | p[2];
    if ((bits >> 8) == pref) atomicAdd(&h[bits & 0xFFu], 1u);
}

// single-thread descending scan: find bin where cumulative count reaches target
__global__ void k_scan(const u32* __restrict__ h, int nbins, u32* __restrict__ p,
                       const int* __restrict__ kptr, int stage) {
    u32 target;
    int outIdx;
    if (stage == 0)      { target = (u32)(*kptr); outIdx = 0; }
    else if (stage == 1) { target = p[1];         outIdx = 2; }
    else                 { target = p[3];         outIdx = 4; }
    if (target == 0u) target = 1u;
    u32 cum = 0, bin = 0, rem = target;
    for (int b = nbins - 1; b >= 0; --b) {
        u32 c = h[b];
        if (cum + c >= target) { bin = (u32)b; rem = target - cum; break; }
        cum += c;
    }
    p[outIdx] = bin;
    p[outIdx + 1] = rem;
}

// mask to exactly k edges (ties resolved by atomic slot counter p[6]),
// then segment-sum selected weights into per-dst-node accumulators.
__global__ void k_mask(float* __restrict__ w, const int* __restrict__ dst,
                       float* __restrict__ inw, u32* __restrict__ p, int E) {
    int e = blockIdx.x * blockDim.x + threadIdx.x;
    if (e >= E) return;
    float sc = w[e];
    u32 bits = __float_as_uint(sc);
    u32 T = (p[0] << 20) | (p[2] << 8) | p[4];
    float wv = 0.f;
    if (bits > T) {
        wv = sc;
    } else if (bits == T) {
        u32 slot = atomicAdd(&p[6], 1u);
        if (slot < p[5]) wv = sc;
    }
    w[e] = wv;
    if (wv != 0.f) atomicAdd(&inw[dst[e]], wv);
}

// ---------------- symmetric degree normalization, interleaved [E,2] output ----------------
__global__ void k_final(const float* __restrict__ wa, const float* __restrict__ wb,
                        const float* __restrict__ inwa, const float* __restrict__ inwb,
                        const int* __restrict__ src, const int* __restrict__ dst,
                        float* __restrict__ out, int E) {
    int e = blockIdx.x * blockDim.x + threadIdx.x;
    if (e >= E) return;
    int s = src[e], d = dst[e];
    float da = rsqrtf(inwa[s] + EPS_) * rsqrtf(inwa[d] + EPS_);
    float db = rsqrtf(inwb[s] + EPS_) * rsqrtf(inwb[d] + EPS_);
    out[2 * (size_t)e]     = wa[e] * da;
    out[2 * (size_t)e + 1] = wb[e] * db;
}

extern "C" void kernel_launch(void* const* d_in, const int* in_sizes, int n_in,
                              void* d_out, int out_size, void* d_ws, size_t ws_size,
                              hipStream_t stream) {
    const float* fa = (const float*)d_in[0];
    const float* fb = (const float*)d_in[1];
    const int* src  = (const int*)d_in[2];
    const int* dst  = (const int*)d_in[3];
    const int* kptr = (const int*)d_in[4];
    int E = in_sizes[2];
    int N = in_sizes[0] / 64;

    // workspace layout (4-byte words)
    float* ws   = (float*)d_ws;
    float* sa   = ws;                      // [E] scores -> clipped weights (mod A)
    float* sb   = ws + (size_t)E;          // [E] scores -> clipped weights (mod B)
    float* na   = sb + (size_t)E;          // [N] node norms (mod A)
    float* nb   = na + N;                  // [N] node norms (mod B)
    float* inwa = nb + N;                  // [N] incoming weight sums (zeroed)
    float* inwb = inwa + N;                // [N]
    const int HSTR = 4096 + 4096 + 256 + 16;
    u32* hbase = (u32*)(inwb + N);
    u32* h1a = hbase;            u32* h2a = h1a + 4096; u32* h3a = h2a + 4096; u32* pa = h3a + 256;
    u32* h1b = hbase + HSTR;     u32* h2b = h1b + 4096; u32* h3b = h2b + 4096; u32* pb = h3b + 256;

    int zeroCount = 2 * N + 2 * HSTR;  // inwa..inwb + both hist/param regions (contiguous)
    k_zero<<<(zeroCount + 255) / 256, 256, 0, stream>>>((u32*)inwa, zeroCount);

    k_norms<<<((N * 32) + 255) / 256, 256, 0, stream>>>(fa, fb, na, nb, N);

    int waves = (E + 15) / 16;
    k_score<<<((waves * 32) + 255) / 256, 256, 0, stream>>>(fa, fb, na, nb, src, dst, sa, sb, E);

    int gb = (E + 255) / 256;
    // modality A top-k + segment sum
    k_hist1<<<gb, 256, 0, stream>>>(sa, h1a, E);
    k_scan<<<1, 1, 0, stream>>>(h1a, 4096, pa, kptr, 0);
    k_hist2<<<gb, 256, 0, stream>>>(sa, pa, h2a, E);
    k_scan<<<1, 1, 0, stream>>>(h2a, 4096, pa, kptr, 1);
    k_hist3<<<gb, 256, 0, stream>>>(sa, pa, h3a, E);
    k_scan<<<1, 1, 0, stream>>>(h3a, 256, pa, kptr, 2);
    k_mask<<<gb, 256, 0, stream>>>(sa, dst, inwa, pa, E);
    // modality B top-k + segment sum
    k_hist1<<<gb, 256, 0, stream>>>(sb, h1b, E);
    k_scan<<<1, 1, 0, stream>>>(h1b, 4096, pb, kptr, 0);
    k_hist2<<<gb, 256, 0, stream>>>(sb, pb, h2b, E);
    k_scan<<<1, 1, 0, stream>>>(h2b, 4096, pb, kptr, 1);
    k_hist3<<<gb, 256, 0, stream>>>(sb, pb, h3b, E);
    k_scan<<<1, 1, 0, stream>>>(h3b, 256, pb, kptr, 2);
    k_mask<<<gb, 256, 0, stream>>>(sb, dst, inwb, pb, E);

    k_final<<<gb, 256, 0, stream>>>(sa, sb, inwa, inwb, src, dst, (float*)d_out, E);
}